// PostProcess_51281909514823
// MI455X (gfx1250) — compile-verified
//
#include <hip/hip_runtime.h>
#include <hip/hip_bf16.h>

#define B_       16
#define N_       25200
#define NC_      80
#define NF_      85
#define MAX_NMS_ 30000
#define MAX_DET_ 300
#define NBINS_   16400
#define CONF_    0.4f
#define IOU_     0.5f
#define MAXWH_   4096.0f

typedef unsigned int u32x4 __attribute__((ext_vector_type(4)));
typedef unsigned int u32x8 __attribute__((ext_vector_type(8)));

// Monotonic bucket key for scores in (CONF_, 1.0): float bits are monotonic
// for positive floats; shift into 16384 buckets.
__device__ __forceinline__ int score_key(float s) {
    unsigned bits = __float_as_uint(s);
    unsigned k = (bits - 0x3E800000u) >> 10;
    return (int)(k < (unsigned)(NBINS_ - 1) ? k : (unsigned)(NBINS_ - 1));
}

// ---------------- init: zero scores / hist / counters ----------------
__global__ void init_kernel(float* __restrict__ scores, int* __restrict__ hist,
                            int* __restrict__ cnt) {
    int g = blockIdx.x * blockDim.x + threadIdx.x;
    if (g < B_ * MAX_NMS_) scores[g] = 0.0f;
    if (g < B_ * NBINS_)   hist[g] = 0;
    if (g < B_)            cnt[g] = 0;
}

// ---------------- pass 1: score-bit histogram per batch ----------------
__global__ void hist_kernel(const float* __restrict__ pred, int* __restrict__ hist) {
    int g = blockIdx.x * blockDim.x + threadIdx.x;
    if (g >= B_ * N_) return;
    int b = g / N_;
    int i = g - b * N_;
    const float* row = pred + ((size_t)b * N_ + i) * NF_;
    __builtin_prefetch(row, 0, 1);   // global_prefetch_b8
    float obj = row[4];
    if (obj <= CONF_) return;
    int* h = hist + b * NBINS_;
    for (int c = 0; c < NC_; ++c) {
        float s = row[5 + c] * obj;
        if (s > CONF_) atomicAdd(&h[score_key(s)], 1);
    }
}

// ---------------- pass 2: find deterministic threshold bucket ----------------
__global__ void thresh_kernel(const int* __restrict__ hist, int* __restrict__ thr) {
    int b = threadIdx.x;
    if (b >= B_) return;
    const int* h = hist + b * NBINS_;
    int cum = 0, T = -1;
    for (int bin = NBINS_ - 1; bin >= 0; --bin) {
        int c = h[bin];
        if (cum + c > MAX_NMS_) { T = bin; break; }
        cum += c;
    }
    thr[b] = T;   // take strictly-greater-than-T keys: count <= MAX_NMS_, deterministic set
}

// ---------------- pass 3: compact candidates ----------------
__global__ void compact_kernel(const float* __restrict__ pred,
                               const int* __restrict__ thr,
                               int* __restrict__ cnt,
                               float* __restrict__ scores,
                               float4* __restrict__ offbox,
                               float* __restrict__ rawbox,
                               float* __restrict__ clsf) {
    int g = blockIdx.x * blockDim.x + threadIdx.x;
    if (g >= B_ * N_) return;
    int b = g / N_;
    int i = g - b * N_;
    const float* row = pred + ((size_t)b * N_ + i) * NF_;
    float obj = row[4];
    if (obj <= CONF_) return;
    float cx = row[0], cy = row[1], w = row[2], h = row[3];
    float x0 = cx - 0.5f * w, y0 = cy - 0.5f * h;
    float x1 = cx + 0.5f * w, y1 = cy + 0.5f * h;
    const int T = thr[b];
    for (int c = 0; c < NC_; ++c) {
        float s = row[5 + c] * obj;
        if (s > CONF_ && score_key(s) > T) {
            int pos = atomicAdd(&cnt[b], 1);
            if (pos < MAX_NMS_) {
                size_t base = (size_t)b * MAX_NMS_ + pos;
                float off = (float)c * MAXWH_;
                offbox[base] = make_float4(x0 + off, y0 + off, x1 + off, y1 + off);
                rawbox[base * 4 + 0] = x0;
                rawbox[base * 4 + 1] = y0;
                rawbox[base * 4 + 2] = x1;
                rawbox[base * 4 + 3] = y1;
                scores[base] = s;
                clsf[base] = (float)c;
            }
        }
    }
}

// ---------------- pass 4: greedy NMS, one workgroup per batch ----------------
__global__ __launch_bounds__(1024, 1)
void nms_kernel(const float* __restrict__ scores_g,
                const float4* __restrict__ offbox,
                const float* __restrict__ rawbox,
                const float* __restrict__ clsf,
                float* __restrict__ out) {
    __shared__ float s_scores[MAX_NMS_];   // 120 KB of the 320 KB WGP LDS
    __shared__ float s_vals[32];
    __shared__ int   s_idxs[32];
    __shared__ float s_win[6];             // box(4), area, value
    __shared__ int   s_widx;

    const int b    = blockIdx.x;
    const int tid  = threadIdx.x;
    const int lane = tid & 31;
    const int wv   = tid >> 5;

    const float*  sg = scores_g + (size_t)b * MAX_NMS_;
    const float4* ob = offbox   + (size_t)b * MAX_NMS_;
    const float*  rb = rawbox   + (size_t)b * MAX_NMS_ * 4;
    const float*  cf = clsf     + (size_t)b * MAX_NMS_;
    float*        o  = out      + (size_t)b * MAX_DET_ * 6;

    // ---- TDM: DMA the 30000-score tile Global -> LDS (wave 0 issues once) ----
    if (wv == 0) {
        unsigned long long ga = (unsigned long long)(const void*)sg;
        // generic shared address: low 32 bits == LDS byte offset (aperture form)
        unsigned lda = (unsigned)(unsigned long long)(const void*)s_scores;
        const unsigned n = MAX_NMS_;
        u32x4 g0;
        g0[0] = 1u;                                            // count=1, user D#
        g0[1] = lda;                                           // lds_addr (bytes)
        g0[2] = (unsigned)(ga & 0xFFFFFFFFull);                // global_addr[31:0]
        g0[3] = ((unsigned)(ga >> 32) & 0x01FFFFFFu)           // global_addr[56:32]
                | (2u << 30);                                  // type = 2 ("image")
        u32x8 g1;
        g1[0] = (2u << 16);                  // workgroup_mask=0, data_size=4B
        g1[1] = (n & 0xFFFFu) << 16;         // tensor_dim0[15:0]
        g1[2] = (n >> 16) | (1u << 16);      // tensor_dim0[31:16], tensor_dim1=1
        g1[3] = (n & 0xFFFFu) << 16;         // tile_dim0 = n
        g1[4] = 1u;                          // tile_dim1 = 1, tile_dim2 = 0
        g1[5] = n;                           // tensor_dim0_stride[31:0]
        g1[6] = (n & 0xFFFFu) << 16;         // stride0 hi=0, stride1 lo
        g1[7] = (n >> 16);                   // stride1 hi
        asm volatile("tensor_load_to_lds %0, %1"
                     :
                     : "s"(g0), "s"(g1)
                     : "memory");
        __builtin_amdgcn_s_wait_tensorcnt(0);
    }
    __syncthreads();

    for (int it = 0; it < MAX_DET_; ++it) {
        // -------- parallel argmax over LDS scores --------
        float best = 0.0f;
        int bix = -1;
        for (int i = tid; i < MAX_NMS_; i += 1024) {
            float v = s_scores[i];
            if (v > best) { best = v; bix = i; }
        }
        for (int off = 16; off > 0; off >>= 1) {
            float ov = __shfl_down(best, off, 32);
            int   oi = __shfl_down(bix,  off, 32);
            if (ov > best) { best = ov; bix = oi; }
        }
        if (lane == 0) { s_vals[wv] = best; s_idxs[wv] = bix; }
        __syncthreads();
        if (wv == 0) {
            float v = s_vals[lane];
            int  ix = s_idxs[lane];
            for (int off = 16; off > 0; off >>= 1) {
                float ov = __shfl_down(v,  off, 32);
                int   oi = __shfl_down(ix, off, 32);
                if (ov > v) { v = ov; ix = oi; }
            }
            if (lane == 0) {
                s_widx  = ix;
                s_win[5] = v;
                if (v > 0.0f && ix >= 0) {
                    float4 wbx = ob[ix];
                    s_win[0] = wbx.x; s_win[1] = wbx.y;
                    s_win[2] = wbx.z; s_win[3] = wbx.w;
                    s_win[4] = (wbx.z - wbx.x) * (wbx.w - wbx.y);
                    s_scores[ix] = 0.0f;       // self-suppress (idx_range == idx)
                }
            }
        }
        __syncthreads();

        const int   ix   = s_widx;
        const float sval = s_win[5];
        const bool  keep = (sval > 0.0f) && (ix >= 0);

        if (tid < 6) {
            float val = 0.0f;
            if (keep) {
                if (tid < 4)       val = rb[(size_t)ix * 4 + tid];
                else if (tid == 4) val = sval;
                else               val = cf[ix];
            }
            o[(size_t)it * 6 + tid] = val;
        }

        if (keep) {
            const float b0 = s_win[0], b1 = s_win[1];
            const float b2 = s_win[2], b3 = s_win[3], ba = s_win[4];
            for (int i = tid; i < MAX_NMS_; i += 1024) {
                float v = s_scores[i];
                if (v <= 0.0f) continue;       // skip dead entries (saves L2 traffic)
                float4 x = ob[i];              // L2-resident float4
                float xx0 = fmaxf(b0, x.x), yy0 = fmaxf(b1, x.y);
                float xx1 = fminf(b2, x.z), yy1 = fminf(b3, x.w);
                float ww = fmaxf(xx1 - xx0, 0.0f);
                float hh = fmaxf(yy1 - yy0, 0.0f);
                float inter = ww * hh;
                float ai = (x.z - x.x) * (x.w - x.y);
                float iou = inter / (ba + ai - inter + 1e-9f);
                if (iou > IOU_) s_scores[i] = 0.0f;
            }
        }
        __syncthreads();
    }
}

extern "C" void kernel_launch(void* const* d_in, const int* in_sizes, int n_in,
                              void* d_out, int out_size, void* d_ws, size_t ws_size,
                              hipStream_t stream) {
    const float* pred = (const float*)d_in[0];
    float* out = (float*)d_out;

    char* ws = (char*)d_ws;
    size_t off = 0;
    float4* offbox = (float4*)(ws + off); off += (size_t)B_ * MAX_NMS_ * sizeof(float4);
    float*  rawbox = (float*) (ws + off); off += (size_t)B_ * MAX_NMS_ * 4 * sizeof(float);
    float*  scores = (float*) (ws + off); off += (size_t)B_ * MAX_NMS_ * sizeof(float);
    float*  clsf   = (float*) (ws + off); off += (size_t)B_ * MAX_NMS_ * sizeof(float);
    int*    hist   = (int*)   (ws + off); off += (size_t)B_ * NBINS_ * sizeof(int);
    int*    thr    = (int*)   (ws + off); off += (size_t)B_ * sizeof(int);
    int*    cnt    = (int*)   (ws + off); off += (size_t)B_ * sizeof(int);
    (void)ws_size; (void)in_sizes; (void)n_in; (void)out_size;

    const int initN = B_ * MAX_NMS_;  // largest region to zero
    init_kernel<<<(initN + 255) / 256, 256, 0, stream>>>(scores, hist, cnt);

    const int anchors = B_ * N_;
    hist_kernel<<<(anchors + 255) / 256, 256, 0, stream>>>(pred, hist);
    thresh_kernel<<<1, 32, 0, stream>>>(hist, thr);
    compact_kernel<<<(anchors + 255) / 256, 256, 0, stream>>>(pred, thr, cnt, scores,
                                                              offbox, rawbox, clsf);
    nms_kernel<<<B_, 1024, 0, stream>>>(scores, offbox, rawbox, clsf, out);
}